// textboxesLoss_38036230373614
// MI455X (gfx1250) — compile-verified
//
#include <hip/hip_runtime.h>
#include <cstdint>

typedef float vf4 __attribute__((ext_vector_type(4)));

// Pointer types matching the builtin's prototype exactly:
// 'int __vector(4) __device__ *' == gcc-vector int4 in target addrspace(1).
typedef int v4i_gcc __attribute__((vector_size(16)));
typedef v4i_gcc __attribute__((address_space(1)))* as1_v4i_ptr;
typedef v4i_gcc __attribute__((address_space(3)))* as3_v4i_ptr;

// CDNA5 async global->LDS path, probe-guarded so a missing/renamed builtin
// falls back to a plain LDS store instead of breaking the compile.
#if defined(__has_builtin)
#if __has_builtin(__builtin_amdgcn_global_load_async_to_lds_b128) && \
    __has_builtin(__builtin_amdgcn_s_wait_asynccnt)
#define ASYNC_LDS_OK 1
#endif
#endif

namespace {
constexpr int Bb  = 4;
constexpr int Nn  = 16384;
constexpr int Kk  = 12;
constexpr int Gg  = 32;
constexpr int NK  = Nn * Kk;        // 196608 boxes per batch
constexpr int ROW = 24 + 4 * Kk;    // 72 floats per out row
constexpr int OFF = 24;             // box data starts at column 24
constexpr int TPB = 256;            // 8 waves (wave32) per block
}

__global__ __launch_bounds__(TPB) void textboxes_iou_kernel(
    const float* __restrict__ outp,   // (B, N, 72)
    const float* __restrict__ dbox,   // (N, 48)
    const float* __restrict__ lines,  // (G, 4)
    float* __restrict__ iou) {        // (B, G, N*K)
  __shared__ vf4  lraw[Gg];
  __shared__ float lx1[Gg], ly1[Gg], lx2[Gg], ly2[Gg], lsb[Gg];

  const int t = threadIdx.x;

  // Stage the 512B lines table into LDS via the CDNA5 async-copy path
  // (tracked by ASYNCcnt -> s_wait_asynccnt).
  if (t < Gg) {
#if ASYNC_LDS_OK
    __builtin_amdgcn_global_load_async_to_lds_b128(
        (as1_v4i_ptr)(uintptr_t)(lines + 4 * t),
        (as3_v4i_ptr)(unsigned)(uintptr_t)&lraw[t],
        /*offset=*/0, /*cpol=*/0);
#else
    lraw[t] = ((const vf4*)lines)[t];
#endif
  }
#if ASYNC_LDS_OK
  __builtin_amdgcn_s_wait_asynccnt(0);
#endif
  __syncthreads();

  if (t < Gg) {
    vf4 L = lraw[t];
    lx1[t] = L.x;
    ly1[t] = L.y;
    lx2[t] = L.z;
    ly2[t] = L.w;
    lsb[t] = (L.z - L.x) * (L.w - L.y);  // SB per line
  }
  __syncthreads();

  // One thread decodes 4 consecutive boxes (quad never crosses an n-row since
  // 12 % 4 == 0) and writes one float4 per line -> 32 NT b128 store streams.
  const int tid = blockIdx.x * TPB + t;          // < B*NK/4 = 196608
  const int b   = tid / (NK / 4);                // uniform per block
  const int nk0 = (tid - b * (NK / 4)) * 4;      // multiple of 4
  const int n   = nk0 / Kk;
  const int k0  = nk0 % Kk;                      // 0, 4, or 8

  const vf4* po = (const vf4*)(outp + (size_t)(b * Nn + n) * ROW + OFF + 4 * k0);
  const vf4* pd = (const vf4*)(dbox + (size_t)n * (4 * Kk) + 4 * k0);

  float px1[4], py1[4], px2[4], py2[4], SA[4];
#pragma unroll
  for (int i = 0; i < 4; ++i) {
    vf4 ob = po[i];  // xb, yb, wb, hb
    vf4 db = pd[i];  // xd, yd, wd, hd
    float x = db.x + db.z * ob.x;
    float y = db.y + db.w * ob.y;
    float w = db.z * __expf(ob.z);
    float h = db.w * __expf(ob.w);
    px1[i] = x - 0.5f * w;
    py1[i] = y - 0.5f * h;
    px2[i] = x + 0.5f * w;
    py2[i] = y + 0.5f * h;
    SA[i]  = (px2[i] - px1[i]) * (py2[i] - py1[i]);
  }

  float* base = iou + (size_t)b * Gg * NK + nk0;
#pragma unroll
  for (int g = 0; g < Gg; ++g) {
    const float gx1 = lx1[g], gy1 = ly1[g], gx2 = lx2[g], gy2 = ly2[g];
    const float SB = lsb[g];
    vf4 v;
#pragma unroll
    for (int i = 0; i < 4; ++i) {
      float W = fminf(px2[i], gx2) - fmaxf(px1[i], gx1);
      float H = fminf(py2[i], gy2) - fmaxf(py1[i], gy1);
      float cross = fmaxf(W, 0.0f) * fmaxf(H, 0.0f);
      float denom = fmaxf(SA[i] + SB - cross, 1e-12f);
      float r = cross * __builtin_amdgcn_rcpf(denom);  // v_rcp_f32 + v_mul
      v[i] = (W > 0.0f && H > 0.0f) ? r : 0.0f;
    }
    // Write-once 100MB output stream: non-temporal so it doesn't evict the
    // reused default_boxes / lines working set from L2.
    __builtin_nontemporal_store(v, (vf4*)(base + (size_t)g * NK));
  }
}

extern "C" void kernel_launch(void* const* d_in, const int* in_sizes, int n_in,
                              void* d_out, int out_size, void* d_ws, size_t ws_size,
                              hipStream_t stream) {
  const float* outp  = (const float*)d_in[0];  // (4, 16384, 72) f32
  const float* dbox  = (const float*)d_in[1];  // (16384, 48) f32
  const float* lines = (const float*)d_in[2];  // (32, 4) f32
  float* iou = (float*)d_out;                  // (4, 32, 196608) f32

  const int total_threads = Bb * (NK / 4);     // 196608
  const int blocks = total_threads / TPB;      // 768
  textboxes_iou_kernel<<<blocks, TPB, 0, stream>>>(outp, dbox, lines, iou);
}